// RelationModule_28260884808186
// MI455X (gfx1250) — compile-verified
//
#include <hip/hip_runtime.h>

typedef __attribute__((ext_vector_type(2))) float v2f;
typedef __attribute__((ext_vector_type(8))) float v8f;

#define B_ 32
#define N_ 2048
#define F_ 32

// D = A(16x4) * B(4x16) + C, all fp32, wave32.
// A layout: M = lane&15, K = vgpr + 2*(lane>>4)
// B layout: N = lane&15, K = vgpr + 2*(lane>>4)
// C/D layout: M = vgpr + 8*(lane>>4), N = lane&15
__device__ __forceinline__ v8f wmma_f32(v2f a, v2f b, v8f c) {
    return __builtin_amdgcn_wmma_f32_16x16x4_f32(
        /*neg_a=*/false, a, /*neg_b=*/false, b,
        /*c_mod=*/(short)0, c, /*reuse_a=*/false, /*reuse_b=*/false);
}

// ---------------------------------------------------------------------------
// Kernel 0: zero the per-batch M accumulators.
// ---------------------------------------------------------------------------
__global__ void k_zero(float* __restrict__ p, int n) {
    int i = blockIdx.x * blockDim.x + threadIdx.x;
    if (i < n) p[i] = 0.0f;
}

// ---------------------------------------------------------------------------
// Pass 1: from x only, compute
//   diag[i] = psi_i . phi_i          (stored, B*N floats)
//   M[b]   += phi[b]^T @ u[b]        (atomic-accumulated, B*F*F floats)
// psi/phi/u live only in registers/LDS. One wave = one 16-row tile.
// ---------------------------------------------------------------------------
__global__ __launch_bounds__(256)
void k_pass1(const float* __restrict__ x,
             const float* __restrict__ w_psi, const float* __restrict__ b_psi,
             const float* __restrict__ w_phi, const float* __restrict__ b_phi,
             const float* __restrict__ w_u,   const float* __restrict__ b_u,
             float* __restrict__ diag, float* __restrict__ M)
{
    __shared__ float Lphi[8][16 * 33];   // per-wave 16x32 tiles, padded stride
    __shared__ float Lu[8][16 * 33];
    __shared__ float Msh[F_ * F_];

    const int b    = blockIdx.y;
    const int wave = threadIdx.x >> 5;
    const int lane = threadIdx.x & 31;
    const int half = lane >> 4;
    const int l16  = lane & 15;
    const int rowBase = blockIdx.x * 128 + wave * 16;

    for (int i = threadIdx.x; i < F_ * F_; i += 256) Msh[i] = 0.0f;

    // A fragments of the 16x32 x-tile (K = feature dim, 8 chunks of 4)
    const float* xrow = x + ((size_t)b * N_ + rowBase + l16) * F_;
    v2f a[8];
#pragma unroll
    for (int kk = 0; kk < 8; ++kk)
        a[kk] = *(const v2f*)(xrow + 4 * kk + 2 * half);

    v8f cpsi[2], cphi[2], cu[2];
#pragma unroll
    for (int nt = 0; nt < 2; ++nt) {
        const int col = nt * 16 + l16;
        v8f accp, accf, accu;
#pragma unroll
        for (int r = 0; r < 8; ++r) {
            accp[r] = b_psi[col];
            accf[r] = b_phi[col];
            accu[r] = b_u[col];
        }
#pragma unroll
        for (int kk = 0; kk < 8; ++kk) {
            const int k0 = 4 * kk + 2 * half;
            v2f bp, bf, bu;
            bp.x = w_psi[k0 * F_ + col]; bp.y = w_psi[(k0 + 1) * F_ + col];
            bf.x = w_phi[k0 * F_ + col]; bf.y = w_phi[(k0 + 1) * F_ + col];
            bu.x = w_u  [k0 * F_ + col]; bu.y = w_u  [(k0 + 1) * F_ + col];
            accp = wmma_f32(a[kk], bp, accp);
            accf = wmma_f32(a[kk], bf, accf);
            accu = wmma_f32(a[kk], bu, accu);
        }
#pragma unroll
        for (int r = 0; r < 8; ++r) accu[r] = fmaxf(accu[r], 0.0f);
        cpsi[nt] = accp; cphi[nt] = accf; cu[nt] = accu;
    }

    // diag[i] = sum_f psi[i,f]*phi[i,f]: reduce across 16 lanes of each half
#pragma unroll
    for (int r = 0; r < 8; ++r) {
        float t = cpsi[0][r] * cphi[0][r] + cpsi[1][r] * cphi[1][r];
        t += __shfl_xor(t, 1, 32);
        t += __shfl_xor(t, 2, 32);
        t += __shfl_xor(t, 4, 32);
        t += __shfl_xor(t, 8, 32);
        if (l16 == 0)
            diag[(size_t)b * N_ + rowBase + r + 8 * half] = t;
    }

    // Stage phi/u tiles to LDS (D layout -> row-major 16x32)
#pragma unroll
    for (int nt = 0; nt < 2; ++nt) {
#pragma unroll
        for (int r = 0; r < 8; ++r) {
            const int li = (r + 8 * half) * 33 + nt * 16 + l16;
            Lphi[wave][li] = cphi[nt][r];
            Lu[wave][li]   = cu[nt][r];
        }
    }
    __syncthreads();   // also covers Msh zeroing

    // M partial: phi^T @ u over this wave's 16 rows (K = local row, 4 chunks)
    v8f acc[2][2];
#pragma unroll
    for (int fa = 0; fa < 2; ++fa)
#pragma unroll
        for (int fb = 0; fb < 2; ++fb)
#pragma unroll
            for (int r = 0; r < 8; ++r) acc[fa][fb][r] = 0.0f;

#pragma unroll
    for (int kk = 0; kk < 4; ++kk) {
        const int k0 = 4 * kk + 2 * half;
        v2f pa[2], ub[2];
#pragma unroll
        for (int t = 0; t < 2; ++t) {
            pa[t].x = Lphi[wave][k0 * 33 + t * 16 + l16];
            pa[t].y = Lphi[wave][(k0 + 1) * 33 + t * 16 + l16];
            ub[t].x = Lu[wave][k0 * 33 + t * 16 + l16];
            ub[t].y = Lu[wave][(k0 + 1) * 33 + t * 16 + l16];
        }
#pragma unroll
        for (int fa = 0; fa < 2; ++fa)
#pragma unroll
            for (int fb = 0; fb < 2; ++fb)
                acc[fa][fb] = wmma_f32(pa[fa], ub[fb], acc[fa][fb]);
    }

#pragma unroll
    for (int fa = 0; fa < 2; ++fa)
#pragma unroll
        for (int fb = 0; fb < 2; ++fb)
#pragma unroll
            for (int r = 0; r < 8; ++r)
                atomicAdd(&Msh[(fa * 16 + r + 8 * half) * F_ + fb * 16 + l16],
                          acc[fa][fb][r]);
    __syncthreads();

    float* Mg = M + (size_t)b * F_ * F_;
    for (int i = threadIdx.x; i < F_ * F_; i += 256)
        atomicAdd(&Mg[i], Msh[i]);
}

// ---------------------------------------------------------------------------
// Kernel 2: per batch, W1 = w_psi @ M  (32x32),  c1 = b_psi @ M (32-vector).
// One wave per batch.
// ---------------------------------------------------------------------------
__global__ __launch_bounds__(32)
void k_fold(const float* __restrict__ w_psi, const float* __restrict__ b_psi,
            const float* __restrict__ M,
            float* __restrict__ W1, float* __restrict__ c1)
{
    const int b    = blockIdx.x;
    const int lane = threadIdx.x;
    const int half = lane >> 4;
    const int l16  = lane & 15;
    const float* Mb = M + (size_t)b * F_ * F_;

    // c1[n] = sum_f b_psi[f] * M[f][n]
    float s = 0.0f;
#pragma unroll
    for (int f = 0; f < F_; ++f) s += b_psi[f] * Mb[f * F_ + lane];
    c1[(size_t)b * F_ + lane] = s;

#pragma unroll
    for (int rt = 0; rt < 2; ++rt) {
        v2f a[8];
#pragma unroll
        for (int kk = 0; kk < 8; ++kk)
            a[kk] = *(const v2f*)(w_psi + (rt * 16 + l16) * F_ + 4 * kk + 2 * half);
#pragma unroll
        for (int nt = 0; nt < 2; ++nt) {
            const int col = nt * 16 + l16;
            v8f acc;
#pragma unroll
            for (int r = 0; r < 8; ++r) acc[r] = 0.0f;
#pragma unroll
            for (int kk = 0; kk < 8; ++kk) {
                const int k0 = 4 * kk + 2 * half;
                v2f bm;
                bm.x = Mb[k0 * F_ + col];
                bm.y = Mb[(k0 + 1) * F_ + col];
                acc = wmma_f32(a[kk], bm, acc);
            }
#pragma unroll
            for (int r = 0; r < 8; ++r)
                W1[(size_t)b * F_ * F_ + (rt * 16 + r + 8 * half) * F_ + col] = acc[r];
        }
    }
}

// ---------------------------------------------------------------------------
// Pass 2: from x only:
//   u = relu(x@w_u + b_u)                 (D layout, registers)
//   S = x@W1[b] + c1[b]   (== psi@M)      (D layout, registers)
//   T = (S - diag*u) / N                  (LDS transpose to A layout)
//   out = T @ w_r + x
// ---------------------------------------------------------------------------
__global__ __launch_bounds__(256)
void k_pass2(const float* __restrict__ x,
             const float* __restrict__ w_u, const float* __restrict__ b_u,
             const float* __restrict__ W1,  const float* __restrict__ c1,
             const float* __restrict__ diag, const float* __restrict__ w_r,
             float* __restrict__ out)
{
    __shared__ float Tsh[8][16 * 33];
    const int b    = blockIdx.y;
    const int wave = threadIdx.x >> 5;
    const int lane = threadIdx.x & 31;
    const int half = lane >> 4;
    const int l16  = lane & 15;
    const int rowBase = blockIdx.x * 128 + wave * 16;

    const float* W1b = W1 + (size_t)b * F_ * F_;
    const float* c1b = c1 + (size_t)b * F_;

    const float* xrow = x + ((size_t)b * N_ + rowBase + l16) * F_;
    v2f a[8];
#pragma unroll
    for (int kk = 0; kk < 8; ++kk)
        a[kk] = *(const v2f*)(xrow + 4 * kk + 2 * half);

    const float invN = 1.0f / (float)N_;
#pragma unroll
    for (int nt = 0; nt < 2; ++nt) {
        const int col = nt * 16 + l16;
        v8f accu, accs;
#pragma unroll
        for (int r = 0; r < 8; ++r) {
            accu[r] = b_u[col];
            accs[r] = c1b[col];
        }
#pragma unroll
        for (int kk = 0; kk < 8; ++kk) {
            const int k0 = 4 * kk + 2 * half;
            v2f bu, bw1;
            bu.x  = w_u[k0 * F_ + col];  bu.y  = w_u[(k0 + 1) * F_ + col];
            bw1.x = W1b[k0 * F_ + col];  bw1.y = W1b[(k0 + 1) * F_ + col];
            accu = wmma_f32(a[kk], bu, accu);
            accs = wmma_f32(a[kk], bw1, accs);
        }
#pragma unroll
        for (int r = 0; r < 8; ++r) {
            const int row = rowBase + r + 8 * half;
            const float uu = fmaxf(accu[r], 0.0f);
            const float t  = (accs[r] - diag[(size_t)b * N_ + row] * uu) * invN;
            Tsh[wave][(r + 8 * half) * 33 + col] = t;
        }
    }
    __syncthreads();

    // T in A layout, out = T @ w_r + x
    v2f at[8];
#pragma unroll
    for (int kk = 0; kk < 8; ++kk) {
        const int k0 = 4 * kk + 2 * half;
        at[kk].x = Tsh[wave][l16 * 33 + k0];
        at[kk].y = Tsh[wave][l16 * 33 + k0 + 1];
    }
#pragma unroll
    for (int nt = 0; nt < 2; ++nt) {
        const int col = nt * 16 + l16;
        v8f acc;
#pragma unroll
        for (int r = 0; r < 8; ++r)   // C init = residual x
            acc[r] = x[((size_t)b * N_ + rowBase + r + 8 * half) * F_ + col];
#pragma unroll
        for (int kk = 0; kk < 8; ++kk) {
            const int k0 = 4 * kk + 2 * half;
            v2f bw;
            bw.x = w_r[k0 * F_ + col];
            bw.y = w_r[(k0 + 1) * F_ + col];
            acc = wmma_f32(at[kk], bw, acc);
        }
#pragma unroll
        for (int r = 0; r < 8; ++r)
            out[((size_t)b * N_ + rowBase + r + 8 * half) * F_ + col] = acc[r];
    }
}

// ---------------------------------------------------------------------------
extern "C" void kernel_launch(void* const* d_in, const int* in_sizes, int n_in,
                              void* d_out, int out_size, void* d_ws, size_t ws_size,
                              hipStream_t stream)
{
    const float* x     = (const float*)d_in[0];
    const float* w_psi = (const float*)d_in[1];
    const float* b_psi = (const float*)d_in[2];
    const float* w_phi = (const float*)d_in[3];
    const float* b_phi = (const float*)d_in[4];
    const float* w_u   = (const float*)d_in[5];
    const float* b_u   = (const float*)d_in[6];
    const float* w_r   = (const float*)d_in[7];
    float* out = (float*)d_out;

    // Workspace (< 1 MB): diag | M | W1 | c1
    float* diag = (float*)d_ws;
    float* M    = diag + (size_t)B_ * N_;
    float* W1   = M + (size_t)B_ * F_ * F_;
    float* c1   = W1 + (size_t)B_ * F_ * F_;

    k_zero<<<(B_ * F_ * F_ + 255) / 256, 256, 0, stream>>>(M, B_ * F_ * F_);
    k_pass1<<<dim3(N_ / 128, B_), 256, 0, stream>>>(
        x, w_psi, b_psi, w_phi, b_phi, w_u, b_u, diag, M);
    k_fold<<<B_, 32, 0, stream>>>(w_psi, b_psi, M, W1, c1);
    k_pass2<<<dim3(N_ / 128, B_), 256, 0, stream>>>(
        x, w_u, b_u, W1, c1, diag, w_r, out);
}